// SSIMLoss_75660143886536
// MI455X (gfx1250) — compile-verified
//
#include <hip/hip_runtime.h>

typedef __attribute__((ext_vector_type(2))) float v2f;
typedef __attribute__((ext_vector_type(8))) float v8f;

#define IMG        512
#define TILE       32          // 32x32 output pixels per workgroup
#define TPI        16          // tiles per image edge (512/32)
#define RAW_ROWS   44          // TILE + 10 halo + 2 K-pad rows (initialized!)
#define RAW_STRIDE 48          // 44 halo cols + 4 pad cols, all zero-filled
#define V_ROWS     32
#define V_STRIDE   48          // cols 0..43 valid for horizontal pass
#define THREADS    128         // 4 waves (wave32)

// 11-tap Gaussian, sigma=1.5, normalized (matches reference within fp32)
__device__ __constant__ float G11[11] = {
    0.00102838f, 0.00759874f, 0.03600077f, 0.10936073f, 0.21300553f,
    0.26601172f,
    0.21300553f, 0.10936073f, 0.03600077f, 0.00759874f, 0.00102838f};

__global__ __launch_bounds__(THREADS) void ssim_tile_kernel(
    const float* __restrict__ img1, const float* __restrict__ img2,
    float* __restrict__ partial)
{
    __shared__ float raw1[RAW_ROWS * RAW_STRIDE];
    __shared__ float raw2[RAW_ROWS * RAW_STRIDE];
    __shared__ float vf[5][V_ROWS * V_STRIDE];
    __shared__ float redbuf[THREADS / 32];

    const int tid   = threadIdx.x;
    const int bx    = blockIdx.x;
    const int image = bx >> 8;          // 256 tiles per image
    const int t     = bx & 255;
    const int gy0   = (t >> 4) * TILE;
    const int gx0   = (t & 15) * TILE;
    const float* p1 = img1 + (size_t)image * IMG * IMG;
    const float* p2 = img2 + (size_t)image * IMG * IMG;

    // ---- Phase 1: stage raw tiles (+halo, zero SAME padding) into LDS ----
    // Every slot of raw1/raw2 (incl. K-pad rows/cols) is written => no NaN
    // bit patterns can reach a WMMA operand (NaN * 0 = NaN would be fatal).
    for (int p = tid; p < RAW_ROWS * RAW_STRIDE; p += THREADS) {
        const int rr = p / RAW_STRIDE, cc = p % RAW_STRIDE;
        const int gy = gy0 + rr - 5, gx = gx0 + cc - 5;
        float a = 0.0f, b = 0.0f;
        if (gy >= 0 && gy < IMG && gx >= 0 && gx < IMG) {
            const int idx = gy * IMG + gx;
            a = p1[idx];
            b = p2[idx];
        }
        raw1[p] = a;
        raw2[p] = b;
    }
    __syncthreads();

    // Lane decomposition shared by both WMMA phases (wave32, EXEC all-ones).
    const int wv    = tid >> 5;
    const int lane  = tid & 31;
    const int n     = lane & 15;           // M (A) or N (B) index within tile
    const int khalf = (lane >> 4) << 1;    // lanes 16..31 hold K+2,K+3

    // ---- Phase 2: vertical blur via WMMA  Out = Wband(16x28) x Data(28x16)
    // A[m][k] = g[k-m] (banded weights, constant per lane, reused for all
    // 5 fields and all tiles). B rows are raw data; the 5 field operands
    // {a, b, a*a, b*b, a*b} are formed in registers from 2 LDS load pairs.
    float av0[7], av1[7];
#pragma unroll
    for (int s = 0; s < 7; ++s) {
        const int k0 = 4 * s + khalf;
        const int d0 = k0 - n, d1 = k0 + 1 - n;
        av0[s] = (d0 >= 0 && d0 <= 10) ? G11[d0] : 0.0f;
        av1[s] = (d1 >= 0 && d1 <= 10) ? G11[d1] : 0.0f;
    }

    // 6 output tiles: rows {0,16} x cols {0,16,32}; wave wv takes tt=wv, wv+4
    for (int tt = wv; tt < 6; tt += 4) {
        const int r0v = (tt / 3) << 4;
        const int c0  = (tt % 3) << 4;
        const int col = c0 + n;

        v8f a1 = {0,0,0,0,0,0,0,0}, a2 = a1, a11 = a1, a22 = a1, a12 = a1;
#pragma unroll
        for (int s = 0; s < 7; ++s) {
            // B VGPR0 holds rows k0 (lanes 0-15) / k0+2 (lanes 16-31);
            // VGPR1 holds rows k0+1 / k0+3.  raw row index = r0v + k.
            const int row0 = (r0v + 4 * s + khalf) * RAW_STRIDE + col;
            const int row1 = row0 + RAW_STRIDE;
            const float x0v = raw1[row0], x1v = raw1[row1];
            const float y0v = raw2[row0], y1v = raw2[row1];
            const v2f A  = {av0[s], av1[s]};
            const v2f B1 = {x0v, x1v};
            const v2f B2 = {y0v, y1v};
            const v2f B3 = {x0v * x0v, x1v * x1v};
            const v2f B4 = {y0v * y0v, y1v * y1v};
            const v2f B5 = {x0v * y0v, x1v * y1v};
            a1  = __builtin_amdgcn_wmma_f32_16x16x4_f32(false, A, false, B1,
                                                        (short)0, a1, false, false);
            a2  = __builtin_amdgcn_wmma_f32_16x16x4_f32(false, A, false, B2,
                                                        (short)0, a2, false, false);
            a11 = __builtin_amdgcn_wmma_f32_16x16x4_f32(false, A, false, B3,
                                                        (short)0, a11, false, false);
            a22 = __builtin_amdgcn_wmma_f32_16x16x4_f32(false, A, false, B4,
                                                        (short)0, a22, false, false);
            a12 = __builtin_amdgcn_wmma_f32_16x16x4_f32(false, A, false, B5,
                                                        (short)0, a12, false, false);
        }
        // Scatter D (VGPR r: rows r0v+r / r0v+r+8, col = c0 + lane%16)
        const int drow = (lane >> 4) << 3;
#pragma unroll
        for (int r = 0; r < 8; ++r) {
            const int o = (r0v + drow + r) * V_STRIDE + col;
            vf[0][o] = a1[r];
            vf[1][o] = a2[r];
            vf[2][o] = a11[r];
            vf[3][o] = a22[r];
            vf[4][o] = a12[r];
        }
    }
    __syncthreads();

    // ---- Phase 3: horizontal blur via WMMA  Out = Data(16x28) x Wband(28x16)
    // B[k][n] = g[k-n] (banded weights, per-lane constants); A rows are the
    // vertically-blurred fields, fetched as 8B-aligned ds_load_b64.
    const int r0 = (wv >> 1) << 4;
    const int x0 = (wv & 1) << 4;
    float bv0[7], bv1[7];
#pragma unroll
    for (int s = 0; s < 7; ++s) {
        const int k0 = 4 * s + khalf;
        const int d0 = k0 - n, d1 = k0 + 1 - n;
        bv0[s] = (d0 >= 0 && d0 <= 10) ? G11[d0] : 0.0f;
        bv1[s] = (d1 >= 0 && d1 <= 10) ? G11[d1] : 0.0f;
    }
    const int abase = (r0 + n) * V_STRIDE + x0 + khalf;

    v8f acc[5];
#pragma unroll
    for (int f = 0; f < 5; ++f) {
        v8f c = {0.f, 0.f, 0.f, 0.f, 0.f, 0.f, 0.f, 0.f};
#pragma unroll
        for (int s = 0; s < 7; ++s) {
            const v2f a = *(const v2f*)&vf[f][abase + 4 * s];
            v2f b;
            b.x = bv0[s];
            b.y = bv1[s];
            c = __builtin_amdgcn_wmma_f32_16x16x4_f32(false, a, false, b,
                                                      (short)0, c, false, false);
        }
        acc[f] = c;
    }

    // ---- Phase 4: pointwise SSIM + reduction (sum is layout-agnostic) ----
    const float C1v = 0.0001f;   // 0.01^2
    const float C2v = 0.0009f;   // 0.03^2
    float sum = 0.0f;
#pragma unroll
    for (int i = 0; i < 8; ++i) {
        const float mu1 = acc[0][i], mu2 = acc[1][i];
        const float m11 = mu1 * mu1, m22 = mu2 * mu2, m12 = mu1 * mu2;
        const float s11 = acc[2][i] - m11;
        const float s22 = acc[3][i] - m22;
        const float s12 = acc[4][i] - m12;
        const float num = (2.0f * m12 + C1v) * (2.0f * s12 + C2v);
        const float den = (m11 + m22 + C1v) * (s11 + s22 + C2v);
        sum += num / den;
    }
    // wave32 reduction
#pragma unroll
    for (int off = 16; off > 0; off >>= 1)
        sum += __shfl_down(sum, off, 32);
    if (lane == 0) redbuf[wv] = sum;
    __syncthreads();
    if (tid == 0)
        partial[bx] = redbuf[0] + redbuf[1] + redbuf[2] + redbuf[3];
}

__global__ __launch_bounds__(256) void ssim_reduce_kernel(
    const float* __restrict__ partial, float* __restrict__ out,
    int n, float scale)
{
    __shared__ float red[8];
    float s = 0.0f;
    for (int i = threadIdx.x; i < n; i += 256) s += partial[i];
#pragma unroll
    for (int off = 16; off > 0; off >>= 1)
        s += __shfl_down(s, off, 32);
    const int wv = threadIdx.x >> 5, lane = threadIdx.x & 31;
    if (lane == 0) red[wv] = s;
    __syncthreads();
    if (threadIdx.x == 0) {
        float tot = 0.0f;
#pragma unroll
        for (int i = 0; i < 8; ++i) tot += red[i];
        out[0] = tot * scale;
    }
}

extern "C" void kernel_launch(void* const* d_in, const int* in_sizes, int n_in,
                              void* d_out, int out_size, void* d_ws, size_t ws_size,
                              hipStream_t stream) {
    (void)in_sizes; (void)n_in; (void)out_size; (void)ws_size;
    const float* img1 = (const float*)d_in[0];
    const float* img2 = (const float*)d_in[1];
    float* out     = (float*)d_out;
    float* partial = (float*)d_ws;           // 8192 floats = 32 KB scratch

    const int nblocks = 32 * TPI * TPI;      // 8192 tiles
    ssim_tile_kernel<<<nblocks, THREADS, 0, stream>>>(img1, img2, partial);
    // N = 32*512*512 = 2^23 exactly
    ssim_reduce_kernel<<<1, 256, 0, stream>>>(partial, out, nblocks,
                                              1.0f / 8388608.0f);
}